// Correlation_28037546508485
// MI455X (gfx1250) — compile-verified
//
#include <hip/hip_runtime.h>

typedef __attribute__((ext_vector_type(2))) float v2f;
typedef __attribute__((ext_vector_type(8))) float v8f;

#define B_DIM   8
#define C_DIM   128
#define H_DIM   96
#define W_DIM   128
#define DISP    4
#define NDISP   9                    // 2*DISP+1
#define CH_STRIDE (H_DIM * W_DIM)    // element stride between channels

// One workgroup = 9 waves (288 threads) handles one (b, y, 16-wide x tile).
// Wave w computes dy = w - 4: a banded GEMM  S = A(16x128) * B(128x24),
// built from 32 V_WMMA_F32_16X16X4_F32 steps per 16-wide N tile.
// Boundary handling: clamped addresses + 0/1 float masks (no EXEC divergence
// in the hot loop; zero-masked B reproduces the reference's zero padding).
__global__ __launch_bounds__(288) void corr_wmma_kernel(
    const float* __restrict__ in1,
    const float* __restrict__ in2,
    float* __restrict__ out) {
  __shared__ float lA[C_DIM * 16];      // in1 tile: [c][m], 8 KB
  __shared__ float lS[NDISP * 16 * 32]; // per-wave score tiles, 18 KB

  const int tid  = threadIdx.x;
  const int lane = tid & 31;
  const int wave = tid >> 5;            // 0..8 -> dy = wave - 4
  const int xb   = blockIdx.x * 16;
  const int y    = blockIdx.y;
  const int b    = blockIdx.z;

  // ---- stage the in1 tile (16 pixels x 128 channels) into LDS ----
  const float* p1 = in1 + (size_t)b * C_DIM * CH_STRIDE + (size_t)y * W_DIM + xb;
  for (int i = tid; i < C_DIM * 16; i += 288) {
    const int c = i >> 4, m = i & 15;
    lA[i] = p1[(size_t)c * CH_STRIDE + m];
  }
  __syncthreads();

  // ---- per-lane fragment coordinates (ISA 16x4 fp32 A/B layout) ----
  const int  mn    = lane & 15;          // M for A, N for B
  const int  khalf = (lane >> 4) << 1;   // lanes 16-31 hold K+2, K+3
  const int  row   = y + wave - DISP;    // in2 row for this wave's dy
  const bool rowOK = (row >= 0) && (row < H_DIM);
  const int  rowc  = row < 0 ? 0 : (row >= H_DIM ? H_DIM - 1 : row);
  const int  col0  = xb - DISP + mn;     // N tile 0: x-4 .. x+11
  const int  col1  = col0 + 16;          // N tile 1: x+12 .. x+27
  const int  col0c = col0 < 0 ? 0 : (col0 >= W_DIM ? W_DIM - 1 : col0);
  const int  col1c = col1 < 0 ? 0 : (col1 >= W_DIM ? W_DIM - 1 : col1);
  const float m0 = (rowOK && col0 >= 0 && col0 < W_DIM) ? 1.0f : 0.0f;
  const float m1 = (rowOK && col1 >= 0 && col1 < W_DIM) ? 1.0f : 0.0f;

  const float* p2row = in2 + (size_t)b * C_DIM * CH_STRIDE + (size_t)rowc * W_DIM;

  v8f acc0 = {};
  v8f acc1 = {};
#pragma unroll 4
  for (int k = 0; k < C_DIM; k += 4) {
    const int c = k + khalf;
    v2f a, b0, b1;
    a.x = lA[c * 16 + mn];
    a.y = lA[(c + 1) * 16 + mn];
    const float* pc = p2row + (size_t)c * CH_STRIDE;
    b0.x = pc[col0c] * m0;
    b0.y = pc[CH_STRIDE + col0c] * m0;
    b1.x = pc[col1c] * m1;
    b1.y = pc[CH_STRIDE + col1c] * m1;
    // 8 args: (neg_a, A, neg_b, B, c_mod, C, reuse_a, reuse_b)
    acc0 = __builtin_amdgcn_wmma_f32_16x16x4_f32(false, a, false, b0,
                                                 (short)0, acc0, false, false);
    acc1 = __builtin_amdgcn_wmma_f32_16x16x4_f32(false, a, false, b1,
                                                 (short)0, acc1, false, false);
  }

  // ---- spill scores: S[m][n], n = 0..31 (C/D layout: VGPR r -> M = r + 8*(lane>=16)) ----
  float* S = lS + wave * (16 * 32);
  const int mbase = (lane >> 4) << 3;
#pragma unroll
  for (int r = 0; r < 8; ++r) {
    S[(mbase + r) * 32 + mn]      = acc0[r];
    S[(mbase + r) * 32 + 16 + mn] = acc1[r];
  }
  __syncthreads();

  // ---- extract diagonal band: out[b][wave*9+dxi][y][xb+m] = S[m][m+dxi] / C ----
  const float invC = 1.0f / (float)C_DIM;
  for (int i = lane; i < NDISP * 16; i += 32) {
    const int dxi = i >> 4;   // 0..8
    const int m   = i & 15;   // 16 contiguous x per dxi -> coalesced stores
    const float v = S[m * 32 + m + dxi] * invC;
    // NT store: output is written once and never re-read; keep L2 for inputs.
    __builtin_nontemporal_store(
        v, &out[(((size_t)b * (NDISP * NDISP) + wave * NDISP + dxi) * H_DIM + y)
                * W_DIM + xb + m]);
  }
}

extern "C" void kernel_launch(void* const* d_in, const int* in_sizes, int n_in,
                              void* d_out, int out_size, void* d_ws, size_t ws_size,
                              hipStream_t stream) {
  (void)in_sizes; (void)n_in; (void)out_size; (void)d_ws; (void)ws_size;
  const float* in1 = (const float*)d_in[0];
  const float* in2 = (const float*)d_in[1];
  float* out = (float*)d_out;
  dim3 grid(W_DIM / 16, H_DIM, B_DIM);   // (8, 96, 8) workgroups
  corr_wmma_kernel<<<grid, 288, 0, stream>>>(in1, in2, out);
}